// RNN_77120432767749
// MI455X (gfx1250) — compile-verified
//
#include <hip/hip_runtime.h>

typedef __attribute__((ext_vector_type(2))) float v2f;
typedef __attribute__((ext_vector_type(8))) float v8f;

#define HSZ 5
#define DSZ 178
#define GSZ 20        // 4*H
#define BSZ 2048
#define TSZ 512
#define WSTRIDE 180   // LDS row stride (D padded to multiple of 4, zero-filled)

// ---------------------------------------------------------------------------
// Phase 1: xg[t][b][g] = sum_d x[b][t][d] * w_ih[g][d]
// One wave per 16-row tile (rows = b*T + t flattened). K looped in chunks of 4
// via V_WMMA_F32_16X16X4_F32; N=20 split into two 16-wide accumulators.
// ---------------------------------------------------------------------------
__global__ __launch_bounds__(256) void lstm_xgemm(const float* __restrict__ x,
                                                  const float* __restrict__ w_ih,
                                                  float* __restrict__ xg) {
    __shared__ float lw[32 * WSTRIDE];   // zero-padded w_ih image: rows 20..31, cols 178..179 = 0
    for (int i = threadIdx.x; i < 32 * WSTRIDE; i += 256) lw[i] = 0.0f;
    __syncthreads();
    for (int i = threadIdx.x; i < GSZ * DSZ; i += 256) {
        int g = i / DSZ, d = i - g * DSZ;
        lw[g * WSTRIDE + d] = w_ih[i];
    }
    __syncthreads();

    const int lane      = threadIdx.x & 31;
    const int waveInBlk = threadIdx.x >> 5;
    const long tile     = (long)blockIdx.x * 8 + waveInBlk;
    const long rowBase  = tile * 16;
    const int  m        = lane & 15;          // M (A rows) / N (B cols)
    const int  koff     = (lane >> 4) << 1;   // lanes 0-15: K=k,k+1 ; lanes 16-31: K=k+2,k+3

    const float* xrow = x + (rowBase + m) * (long)DSZ;
    const float* bw0  = &lw[m * WSTRIDE];          // B columns 0..15  -> gates 0..15
    const float* bw1  = &lw[(m + 16) * WSTRIDE];   // B columns 16..31 -> gates 16..19 (+pad)

    v8f c0 = {}, c1 = {};
    #pragma unroll 4
    for (int k = 0; k < 176; k += 4) {
        v2f a, b0, b1;
        a.x  = xrow[k + koff];     a.y  = xrow[k + koff + 1];
        b0.x = bw0 [k + koff];     b0.y = bw0 [k + koff + 1];
        b1.x = bw1 [k + koff];     b1.y = bw1 [k + koff + 1];
        c0 = __builtin_amdgcn_wmma_f32_16x16x4_f32(false, a, false, b0, (short)0, c0, false, false);
        c1 = __builtin_amdgcn_wmma_f32_16x16x4_f32(false, a, false, b1, (short)0, c1, false, false);
    }
    {   // remainder chunk K=176..179 (D=178): zero-pad K>=178 (A via predicate, B via LDS pad)
        v2f a = {}, b0, b1;
        if (koff == 0) { a.x = xrow[176]; a.y = xrow[177]; }
        b0.x = bw0[176 + koff];  b0.y = bw0[176 + koff + 1];
        b1.x = bw1[176 + koff];  b1.y = bw1[176 + koff + 1];
        c0 = __builtin_amdgcn_wmma_f32_16x16x4_f32(false, a, false, b0, (short)0, c0, false, false);
        c1 = __builtin_amdgcn_wmma_f32_16x16x4_f32(false, a, false, b1, (short)0, c1, false, false);
    }

    // C/D layout: VGPR v holds M = v + (lane<16 ? 0 : 8), N = lane&15.
    // Scatter to (t, B, 20) layout for coalesced phase-2 reads.
    const int nc = lane & 15;
    const int mb = (lane < 16) ? 0 : 8;
    #pragma unroll
    for (int v = 0; v < 8; ++v) {
        long r  = rowBase + (long)(mb + v);
        int  bb = (int)(r >> 9);      // T = 512
        int  tt = (int)(r & 511);
        float* dst = xg + ((long)tt * BSZ + bb) * GSZ;
        dst[nc] = c0[v];
        if (nc < 4) dst[16 + nc] = c1[v];
    }
}

// ---------------------------------------------------------------------------
// Phase 2: sequential LSTM scan over T, one thread per batch row, h/c in
// registers, recurrent weights broadcast from LDS, next-step gate prefetch.
// FC + ReLU epilogue folded in.
// ---------------------------------------------------------------------------
__global__ __launch_bounds__(256) void lstm_scan(const float* __restrict__ xg,
                                                 const float* __restrict__ w_hh,
                                                 const float* __restrict__ b_ih,
                                                 const float* __restrict__ b_hh,
                                                 const float* __restrict__ w_fc,
                                                 const float* __restrict__ b_fc,
                                                 float* __restrict__ out) {
    __shared__ float sw[GSZ * HSZ];
    __shared__ float sb[GSZ];
    __shared__ float sfw[HSZ * HSZ];
    __shared__ float sfb[HSZ];
    const int tid = threadIdx.x;
    if (tid < GSZ * HSZ) sw[tid] = w_hh[tid];
    if (tid < GSZ)       sb[tid] = b_ih[tid] + b_hh[tid];
    if (tid >= 128 && tid < 128 + HSZ * HSZ) sfw[tid - 128] = w_fc[tid - 128];
    if (tid >= 160 && tid < 160 + HSZ)       sfb[tid - 160] = b_fc[tid - 160];
    __syncthreads();

    const int b = blockIdx.x * 256 + tid;
    float h[HSZ] = {0, 0, 0, 0, 0};
    float c[HSZ] = {0, 0, 0, 0, 0};

    float4 q[5];
    {
        const float4* gp = (const float4*)(xg + (long)b * GSZ);
        #pragma unroll
        for (int i = 0; i < 5; ++i) q[i] = gp[i];
    }

    for (int t = 0; t < TSZ; ++t) {
        float4 qn[5];
        if (t + 1 < TSZ) {   // prefetch next step under this step's dependent chain
            const float4* gp = (const float4*)(xg + ((long)(t + 1) * BSZ + b) * GSZ);
            #pragma unroll
            for (int i = 0; i < 5; ++i) qn[i] = gp[i];
        }

        float g[GSZ] = {q[0].x, q[0].y, q[0].z, q[0].w,
                        q[1].x, q[1].y, q[1].z, q[1].w,
                        q[2].x, q[2].y, q[2].z, q[2].w,
                        q[3].x, q[3].y, q[3].z, q[3].w,
                        q[4].x, q[4].y, q[4].z, q[4].w};
        #pragma unroll
        for (int j = 0; j < GSZ; ++j) {
            float acc = g[j] + sb[j];
            #pragma unroll
            for (int k2 = 0; k2 < HSZ; ++k2) acc += h[k2] * sw[j * HSZ + k2];
            g[j] = acc;
        }
        #pragma unroll
        for (int k2 = 0; k2 < HSZ; ++k2) {
            float ig = 1.0f / (1.0f + __expf(-g[k2]));        // i
            float fg = 1.0f / (1.0f + __expf(-g[5 + k2]));    // f
            float gg = tanhf(g[10 + k2]);                     // g
            float og = 1.0f / (1.0f + __expf(-g[15 + k2]));   // o
            c[k2] = fg * c[k2] + ig * gg;
            h[k2] = og * tanhf(c[k2]);
        }
        #pragma unroll
        for (int i = 0; i < 5; ++i) q[i] = qn[i];
    }

    // FC + ReLU epilogue: out[b][j] = b_fc[j] + sum_k relu(h[k]) * w_fc[j][k]
    #pragma unroll
    for (int j = 0; j < HSZ; ++j) {
        float acc = sfb[j];
        #pragma unroll
        for (int k2 = 0; k2 < HSZ; ++k2) acc += fmaxf(h[k2], 0.0f) * sfw[j * HSZ + k2];
        out[(long)b * HSZ + j] = acc;
    }
}

extern "C" void kernel_launch(void* const* d_in, const int* in_sizes, int n_in,
                              void* d_out, int out_size, void* d_ws, size_t ws_size,
                              hipStream_t stream) {
    (void)in_sizes; (void)n_in; (void)out_size; (void)ws_size;
    const float* x    = (const float*)d_in[0];
    const float* w_ih = (const float*)d_in[1];
    const float* w_hh = (const float*)d_in[2];
    const float* b_ih = (const float*)d_in[3];
    const float* b_hh = (const float*)d_in[4];
    const float* w_fc = (const float*)d_in[5];
    const float* b_fc = (const float*)d_in[6];
    float* out = (float*)d_out;
    float* xg  = (float*)d_ws;   // needs T*B*4H*4 = 83,886,080 bytes

    const long tiles  = (long)BSZ * TSZ / 16;   // 65536 waves
    const int  blocks = (int)(tiles / 8);       // 8 waves (256 threads) per block

    lstm_xgemm<<<blocks, 256, 0, stream>>>(x, w_ih, xg);
    lstm_scan<<<BSZ / 256, 256, 0, stream>>>(xg, w_hh, b_ih, b_hh, w_fc, b_fc, out);
}